// HierarchicalRGB_field_26766236188897
// MI455X (gfx1250) — compile-verified
//
#include <hip/hip_runtime.h>
#include <stdint.h>

#define BLOCK  256
#define CHUNKS 8

// Low 32 bits of a generic pointer to a __shared__ object == raw LDS byte offset
// (addrspace(3) -> flat keeps the LDS offset in addr[31:0] per the aperture rules).
__device__ __forceinline__ uint32_t lds_off(const void* p) {
  return (uint32_t)(uintptr_t)p;
}

// CDNA5 async copy: global memory -> LDS, tracked by ASYNCcnt (no VGPR data path).
__device__ __forceinline__ void async_ld_b64(uint32_t lds_addr, const void* gaddr) {
  asm volatile("global_load_async_to_lds_b64 %0, %1, off"
               :: "v"(lds_addr), "v"(gaddr)
               : "memory");
}
__device__ __forceinline__ void wait_async_le1() {
  asm volatile("s_wait_asynccnt 0x1" ::: "memory");
}
__device__ __forceinline__ void wait_async_0() {
  asm volatile("s_wait_asynccnt 0x0" ::: "memory");
}

__global__ __launch_bounds__(BLOCK)
void hier_rgb_kernel(const float2* __restrict__ uv,
                     const float*  __restrict__ tex0,
                     const float*  __restrict__ tex1,
                     const float*  __restrict__ tex2,
                     const float*  __restrict__ tex3,
                     float*        __restrict__ out,
                     int n)
{
  __shared__ float2 s_uv[2 * BLOCK];

  const int tid  = threadIdx.x;
  const int base = blockIdx.x * (BLOCK * CHUNKS);

  // ---- prologue: async-stage chunk 0 into LDS buffer 0 ----
  {
    int g0 = base + tid;
    g0 = g0 < n ? g0 : (n - 1);
    async_ld_b64(lds_off(&s_uv[tid]), (const void*)(uv + g0));
  }

  const float* tptr[4] = {tex0, tex1, tex2, tex3};

  #pragma unroll 1
  for (int i = 0; i < CHUNKS; ++i) {
    const int cur = i & 1;

    // ---- pipeline: kick off chunk i+1, then wait for chunk i ----
    if (i + 1 < CHUNKS) {
      const int nxt = (i + 1) & 1;
      int gn = base + (i + 1) * BLOCK + tid;
      gn = gn < n ? gn : (n - 1);
      async_ld_b64(lds_off(&s_uv[nxt * BLOCK + tid]), (const void*)(uv + gn));
      wait_async_le1();            // allow the just-issued load to stay in flight
    } else {
      wait_async_0();
    }

    const float2 p = s_uv[cur * BLOCK + tid];
    const float u = p.x;
    const float v = p.y;

    // ---- phase 1: all 16 texel addresses + weights (res is compile-time) ----
    const int rr[4] = {2048, 1024, 512, 256};
    float wxa[4], wya[4];
    const float* pk[16];
    #pragma unroll
    for (int t = 0; t < 4; ++t) {
      const int   r  = rr[t];
      const float fr = (float)(r - 1);
      const float fx = u * fr;
      const float fy = v * fr;
      const float x0f = floorf(fx);
      const float y0f = floorf(fy);
      wxa[t] = fx - x0f;
      wya[t] = fy - y0f;
      int x0 = (int)x0f;
      int y0 = (int)y0f;
      x0 = min(max(x0, 0), r - 1);
      y0 = min(max(y0, 0), r - 1);
      const int x1 = min(x0 + 1, r - 1);
      const int y1 = min(y0 + 1, r - 1);
      const float* tp = tptr[t];
      const int row0 = y0 * r;
      const int row1 = y1 * r;
      pk[t * 4 + 0] = tp + 3 * (row0 + x0);
      pk[t * 4 + 1] = tp + 3 * (row0 + x1);
      pk[t * 4 + 2] = tp + 3 * (row1 + x0);
      pk[t * 4 + 3] = tp + 3 * (row1 + x1);
    }

    // ---- phase 2: issue all 48 gather loads back-to-back (max MLP) ----
    float tv[16][3];
    #pragma unroll
    for (int k = 0; k < 16; ++k) {
      tv[k][0] = pk[k][0];
      tv[k][1] = pk[k][1];
      tv[k][2] = pk[k][2];
    }

    // ---- phase 3: bilinear blend, faithful to reference FMA order ----
    float o0 = 0.f, o1 = 0.f, o2 = 0.f;
    #pragma unroll
    for (int t = 0; t < 4; ++t) {
      const float wx  = wxa[t];
      const float wy  = wya[t];
      const float iwx = 1.0f - wx;
      const float iwy = 1.0f - wy;
      {
        const float top = tv[t*4+0][0] * iwx + tv[t*4+1][0] * wx;
        const float bot = tv[t*4+2][0] * iwx + tv[t*4+3][0] * wx;
        o0 += top * iwy + bot * wy;
      }
      {
        const float top = tv[t*4+0][1] * iwx + tv[t*4+1][1] * wx;
        const float bot = tv[t*4+2][1] * iwx + tv[t*4+3][1] * wx;
        o1 += top * iwy + bot * wy;
      }
      {
        const float top = tv[t*4+0][2] * iwx + tv[t*4+1][2] * wx;
        const float bot = tv[t*4+2][2] * iwx + tv[t*4+3][2] * wx;
        o2 += top * iwy + bot * wy;
      }
    }

    const int idx = base + i * BLOCK + tid;
    if (idx < n) {
      float* op = out + idx * 3;
      op[0] = o0;
      op[1] = o1;
      op[2] = o2;
    }
  }
}

extern "C" void kernel_launch(void* const* d_in, const int* in_sizes, int n_in,
                              void* d_out, int out_size, void* d_ws, size_t ws_size,
                              hipStream_t stream) {
  const float2* uv = (const float2*)d_in[0];
  const float*  t0 = (const float*)d_in[1];
  const float*  t1 = (const float*)d_in[2];
  const float*  t2 = (const float*)d_in[3];
  const float*  t3 = (const float*)d_in[4];
  float* out = (float*)d_out;

  const int n = in_sizes[0] / 2;                 // number of points
  const int per_block = BLOCK * CHUNKS;          // 2048 points per block
  const int blocks = (n + per_block - 1) / per_block;   // 1024 for N=2M

  hier_rgb_kernel<<<blocks, BLOCK, 0, stream>>>(uv, t0, t1, t2, t3, out, n);
}